// GCN_59897613910372
// MI455X (gfx1250) — compile-verified
//
#include <hip/hip_runtime.h>
#include <hip/hip_bf16.h>

typedef __attribute__((ext_vector_type(2))) float v2f;
typedef __attribute__((ext_vector_type(4))) float v4f;
typedef __attribute__((ext_vector_type(8))) float v8f;
typedef __attribute__((ext_vector_type(4))) int   v4i;

#define N_NODES 160000
#define N_EDGES 5120000
#define HID 16

// LDS tile geometry for the WMMA GEMMs
#define KT 32          // K-extent of one staged tile
#define ST 36          // padded LDS row stride (dwords) -> conflict-free frag reads

// ---------------------------------------------------------------------------
// CDNA5 async global->LDS copy (ASYNCcnt-tracked). Fallback: sync ld/ds_store.
// Builtin signature (probe-confirmed by round-2 diagnostic):
//   (v4i addrspace(1)* gsrc, v4i addrspace(3)* ldst, imm offset, imm cpol)
// ---------------------------------------------------------------------------
#if __has_builtin(__builtin_amdgcn_global_load_async_to_lds_b128)
#define HAVE_ASYNC_LDS 1
__device__ __forceinline__ void async_copy16(const float* gsrc, float* ldst) {
    __builtin_amdgcn_global_load_async_to_lds_b128(
        (__attribute__((address_space(1))) v4i*)gsrc,
        (__attribute__((address_space(3))) v4i*)ldst,
        0, 0);
}
#else
#define HAVE_ASYNC_LDS 0
__device__ __forceinline__ void async_copy16(const float* gsrc, float* ldst) {
    *(v4f*)ldst = *(const v4f*)gsrc;   // global_load_b128 + ds_store_b128
}
#endif

__device__ __forceinline__ void wait_async_le8() {
#if HAVE_ASYNC_LDS
#if __has_builtin(__builtin_amdgcn_s_wait_asynccnt)
    __builtin_amdgcn_s_wait_asynccnt(8);
#else
    asm volatile("s_wait_asynccnt 0x8" ::: "memory");
#endif
#endif
    asm volatile("" ::: "memory");
}

__device__ __forceinline__ void wait_async_le0() {
#if HAVE_ASYNC_LDS
#if __has_builtin(__builtin_amdgcn_s_wait_asynccnt)
    __builtin_amdgcn_s_wait_asynccnt(0);
#else
    asm volatile("s_wait_asynccnt 0x0" ::: "memory");
#endif
#endif
    asm volatile("" ::: "memory");
}

// ---------------------------------------------------------------------------
// Scalar edge aggregation: agg[tgt[e]] += val[src[e]] * ew[e]
// Used for BOTH GraphConv layers (layer 2 collapsed to scalar via linearity).
// ---------------------------------------------------------------------------
__global__ void edge_agg_kernel(float* __restrict__ agg,
                                const float* __restrict__ val,
                                const int* __restrict__ src,
                                const int* __restrict__ tgt,
                                const float* __restrict__ ew, int E) {
    int e = blockIdx.x * blockDim.x + threadIdx.x;
    if (e < E) {
        float m = val[src[e]] * ew[e];
        unsafeAtomicAdd(&agg[tgt[e]], m);   // -> global_atomic_add_f32 (no return)
    }
}

// ---------------------------------------------------------------------------
// Fused per-node layer-1 update + layer-2 projections.
//   h_k   = relu(agg1[n]*Wr1[k] + x[n]*Ws1[k] + br1[k])
//   s[n]  = sum_k h_k * Wr2[k]   (message scalar for layer 2)
//   r[n]  = sum_k h_k * Ws2[k]   (root term for layer 2)
// ---------------------------------------------------------------------------
__global__ void node_update_kernel(float* __restrict__ s, float* __restrict__ r,
                                   const float* __restrict__ agg1,
                                   const float* __restrict__ x,
                                   const float* __restrict__ Wr1,
                                   const float* __restrict__ br1,
                                   const float* __restrict__ Ws1,
                                   const float* __restrict__ Wr2,
                                   const float* __restrict__ Ws2, int N) {
    int n = blockIdx.x * blockDim.x + threadIdx.x;
    if (n >= N) return;
    float a = agg1[n];
    float xn = x[n];
    float sv = 0.f, rv = 0.f;
#pragma unroll
    for (int k = 0; k < HID; ++k) {
        float h = fmaxf(a * Wr1[k] + xn * Ws1[k] + br1[k], 0.f);
        sv += h * Wr2[k];
        rv += h * Ws2[k];
    }
    s[n] = sv;
    r[n] = rv;
}

// ---------------------------------------------------------------------------
// Build padded MLP input g[16][20000]; rows 0..7 real, rows 8..15 zero.
// ---------------------------------------------------------------------------
__global__ void build_g_kernel(float* __restrict__ g,
                               const float* __restrict__ agg2,
                               const float* __restrict__ r,
                               const float* __restrict__ br2) {
    int i = blockIdx.x * blockDim.x + threadIdx.x;   // < 16*20000
    g[i] = (i < N_NODES) ? (agg2[i] + br2[0] + r[i]) : 0.0f;
}

// ---------------------------------------------------------------------------
// WMMA f32 GEMM, LDS-staged + double-buffered async copies:
//   out[16][N] += A[16][K] * W^T, W row-major [N][K].
// One wave per 16x16 output tile per K-chunk of KC (KC % KT == 0).
// Staging: per tile, 4 async B128 instrs/matrix; each instr = 4 rows x
//   (8 lanes x 16B) = 128B line-aligned contiguous segments -> minimal HBM
//   line stream for the 320MB W1 read.
// Frag reads from LDS (stride ST=36 dwords): banks 36*row+2*half+{0,1} mod 64
//   are all distinct -> conflict-free ds_load_b64.
// ---------------------------------------------------------------------------
__global__ void gemm_wmma_kernel(const float* __restrict__ A,
                                 const float* __restrict__ W,
                                 float* __restrict__ out,
                                 int K, int N, int KC) {
    __shared__ float lsA[2][16 * ST];
    __shared__ float lsB[2][16 * ST];

    const int lane = threadIdx.x;          // 0..31, all active (EXEC all-1s)
    const int row  = lane & 15;
    const int half = lane >> 4;
    const int jbase = blockIdx.x * 16;
    const long kbase = (long)blockIdx.y * KC;

    const float* Ab = A + kbase;
    const float* Wb = W + (long)jbase * K + kbase;

    // staging lane mapping: 4 rows x 8 lanes x 16B per instruction
    const int srow = lane >> 3;            // 0..3
    const int scol = (lane & 7) * 4;       // 0,4,...,28

    const int nchunks = KC / KT;

    // prologue: stage chunk 0 into buffer 0
#pragma unroll
    for (int i = 0; i < 4; ++i) {
        int rr = 4 * i + srow;
        async_copy16(Ab + (long)rr * K + scol, &lsA[0][rr * ST + scol]);
        async_copy16(Wb + (long)rr * K + scol, &lsB[0][rr * ST + scol]);
    }

    v8f c = {};
    for (int ci = 0; ci < nchunks; ++ci) {
        const int buf = ci & 1;
        if (ci + 1 < nchunks) {
            const long koff = (long)(ci + 1) * KT;
#pragma unroll
            for (int i = 0; i < 4; ++i) {
                int rr = 4 * i + srow;
                async_copy16(Ab + (long)rr * K + koff + scol,
                             &lsA[buf ^ 1][rr * ST + scol]);
                async_copy16(Wb + (long)rr * K + koff + scol,
                             &lsB[buf ^ 1][rr * ST + scol]);
            }
            wait_async_le8();   // current tile (older 8 async ops) resident
        } else {
            wait_async_le0();
        }

        const float* la = &lsA[buf][row * ST + 2 * half];
        const float* lb = &lsB[buf][row * ST + 2 * half];
#pragma unroll
        for (int k = 0; k < KT; k += 4) {
            v2f a = *(const v2f*)(la + k);
            v2f b = *(const v2f*)(lb + k);
            // (neg_a, A, neg_b, B, c_mod, C, reuse_a, reuse_b)
            c = __builtin_amdgcn_wmma_f32_16x16x4_f32(
                    false, a, false, b, (short)0, c, false, false);
        }
    }

    // C/D layout: vgpr rr, lane -> (M = 8*half + rr, N = jbase + row)
    const int m0 = half * 8;
    float* op = out + jbase + row;
#pragma unroll
    for (int rr = 0; rr < 8; ++rr)
        unsafeAtomicAdd(op + (long)(m0 + rr) * N, c[rr]);
}

// ---------------------------------------------------------------------------
// act[16][N] = relu(out + bias) on rows 0..7; rows 8..15 forced to zero.
// ---------------------------------------------------------------------------
__global__ void bias_relu_kernel(float* __restrict__ act,
                                 const float* __restrict__ out,
                                 const float* __restrict__ b, int N) {
    int i = blockIdx.x * blockDim.x + threadIdx.x;
    if (i < 16 * N) {
        int m = i / N;
        int j = i - m * N;
        act[i] = (m < 8) ? fmaxf(out[i] + b[j], 0.f) : 0.f;
    }
}

// ---------------------------------------------------------------------------
// Final tiny layer [8,160] x [160,10] + bias + log_softmax -> d_out[80].
// ---------------------------------------------------------------------------
__global__ void head_kernel(const float* __restrict__ act3,
                            const float* __restrict__ W4,
                            const float* __restrict__ b4,
                            float* __restrict__ out) {
    __shared__ float lg[8][10];
    int t = threadIdx.x;
    if (t < 80) {
        int m = t / 10, j = t % 10;
        float acc = b4[j];
        for (int k = 0; k < 160; ++k)
            acc += act3[m * 160 + k] * W4[j * 160 + k];
        lg[m][j] = acc;
    }
    __syncthreads();
    if (t < 8) {
        float mx = lg[t][0];
        for (int j = 1; j < 10; ++j) mx = fmaxf(mx, lg[t][j]);
        float se = 0.f;
        for (int j = 0; j < 10; ++j) se += expf(lg[t][j] - mx);
        float ls = logf(se);
        for (int j = 0; j < 10; ++j) out[t * 10 + j] = lg[t][j] - mx - ls;
    }
}

// ---------------------------------------------------------------------------
extern "C" void kernel_launch(void* const* d_in, const int* in_sizes, int n_in,
                              void* d_out, int out_size, void* d_ws, size_t ws_size,
                              hipStream_t stream) {
    const float* x   = (const float*)d_in[0];
    const float* ew  = (const float*)d_in[1];
    const float* Wr1 = (const float*)d_in[2];
    const float* br1 = (const float*)d_in[3];
    const float* Ws1 = (const float*)d_in[4];
    const float* Wr2 = (const float*)d_in[5];
    const float* br2 = (const float*)d_in[6];
    const float* Ws2 = (const float*)d_in[7];
    const float* W1  = (const float*)d_in[8];
    const float* b1  = (const float*)d_in[9];
    const float* W2  = (const float*)d_in[10];
    const float* b2  = (const float*)d_in[11];
    const float* W3  = (const float*)d_in[12];
    const float* b3  = (const float*)d_in[13];
    const float* W4  = (const float*)d_in[14];
    const float* b4  = (const float*)d_in[15];
    const int*   ei  = (const int*)d_in[16];   // [2][E]: row0=src, row1=tgt
    float* out = (float*)d_out;

    // Workspace layout (float element offsets); zero-init region first.
    float* ws   = (float*)d_ws;
    float* agg1 = ws + 0;         // 160000
    float* agg2 = ws + 160000;    // 160000
    float* out1 = ws + 320000;    // 16*4000 = 64000
    float* out2 = ws + 384000;    // 16*800  = 12800
    float* out3 = ws + 396800;    // 16*160  = 2560   -> zero region ends 399360
    float* sbuf = ws + 399360;    // 160000
    float* rbuf = ws + 559360;    // 160000
    float* gbuf = ws + 719360;    // 16*20000 = 320000
    float* act1 = ws + 1039360;   // 64000
    float* act2 = ws + 1103360;   // 12800
    float* act3 = ws + 1116160;   // 2560   (total 1118720 floats ~ 4.5 MB)

    (void)hipMemsetAsync(ws, 0, 399360 * sizeof(float), stream);

    // GraphConv 1 (scalar aggregation) + fused node update
    edge_agg_kernel<<<N_EDGES / 256, 256, 0, stream>>>(agg1, x, ei, ei + N_EDGES, ew, N_EDGES);
    node_update_kernel<<<N_NODES / 256, 256, 0, stream>>>(sbuf, rbuf, agg1, x,
                                                          Wr1, br1, Ws1, Wr2, Ws2, N_NODES);
    // GraphConv 2 collapsed to scalar aggregation of s
    edge_agg_kernel<<<N_EDGES / 256, 256, 0, stream>>>(agg2, sbuf, ei, ei + N_EDGES, ew, N_EDGES);
    build_g_kernel<<<(16 * 20000) / 256, 256, 0, stream>>>(gbuf, agg2, rbuf, br2);

    // MLP funnel via WMMA f32 16x16x4, async-LDS staged (batch padded 8->16)
    gemm_wmma_kernel<<<dim3(250, 25), 32, 0, stream>>>(gbuf, W1, out1, 20000, 4000, 800);
    bias_relu_kernel<<<(16 * 4000) / 256, 256, 0, stream>>>(act1, out1, b1, 4000);

    gemm_wmma_kernel<<<dim3(50, 25), 32, 0, stream>>>(act1, W2, out2, 4000, 800, 160);
    bias_relu_kernel<<<(16 * 800) / 256, 256, 0, stream>>>(act2, out2, b2, 800);

    gemm_wmma_kernel<<<dim3(10, 5), 32, 0, stream>>>(act2, W3, out3, 800, 160, 160);
    bias_relu_kernel<<<10, 256, 0, stream>>>(act3, out3, b3, 160);

    head_kernel<<<1, 128, 0, stream>>>(act3, W4, b4, out);
}